// Model_24086176596107
// MI455X (gfx1250) — compile-verified
//
#include <hip/hip_runtime.h>

typedef __attribute__((ext_vector_type(2))) float v2f;
typedef __attribute__((ext_vector_type(8))) float v8f;

#define N_QUBITS 12
#define N_LAYERS 4
#define DIM      4096   // 2^12
#define BATCH    256

// ---------- complex helpers ----------
__device__ __forceinline__ float2 cmul(float2 a, float2 b) {
    return make_float2(a.x * b.x - a.y * b.y, a.x * b.y + a.y * b.x);
}
__device__ __forceinline__ float2 cadd(float2 a, float2 b) {
    return make_float2(a.x + b.x, a.y + b.y);
}
__device__ __forceinline__ float2 cscale(float s, float2 a) {
    return make_float2(s * a.x, s * a.y);
}

struct C2x2 { float2 m00, m01, m10, m11; };

// M <- RX(t) * M ;  RX = [[c, -i s], [-i s, c]]
__device__ __forceinline__ void lmul_rx(float t, C2x2& M) {
    float c = cosf(0.5f * t), s = sinf(0.5f * t);
    float2 is = make_float2(0.f, -s);
    float2 n00 = cadd(cscale(c, M.m00), cmul(is, M.m10));
    float2 n01 = cadd(cscale(c, M.m01), cmul(is, M.m11));
    float2 n10 = cadd(cmul(is, M.m00), cscale(c, M.m10));
    float2 n11 = cadd(cmul(is, M.m01), cscale(c, M.m11));
    M.m00 = n00; M.m01 = n01; M.m10 = n10; M.m11 = n11;
}
// M <- RZ(t) * M ;  RZ = diag(e^{-it/2}, e^{+it/2})
__device__ __forceinline__ void lmul_rz(float t, C2x2& M) {
    float2 e  = make_float2(cosf(0.5f * t), -sinf(0.5f * t));
    float2 ec = make_float2(e.x, -e.y);
    M.m00 = cmul(e,  M.m00); M.m01 = cmul(e,  M.m01);
    M.m10 = cmul(ec, M.m10); M.m11 = cmul(ec, M.m11);
}

// LDS address of state amplitude whose group-g nibble equals m and whose
// remaining 8 bits are packed into column index c (0..255).
//   g=0: nibble = bits 11..8   -> idx = (m<<8) | c
//   g=1: nibble = bits  7..4   -> idx = (c_hi<<8) | (m<<4) | c_lo
//   g=2: nibble = bits  3..0   -> idx = (c<<4) | m
__device__ __forceinline__ int gaddr(int g, int m, int c) {
    if (g == 0) return (m << 8) | c;
    if (g == 1) return ((c & ~15) << 4) | (m << 4) | (c & 15);
    return (c << 4) | m;
}

__global__ __launch_bounds__(256) void qsim_hea_kernel(const float* __restrict__ x,
                                                       const float* __restrict__ params,
                                                       float* __restrict__ out) {
    __shared__ float  sre[DIM];          // 16 KB  state (real)
    __shared__ float  sim_[DIM];         // 16 KB  state (imag)
    __shared__ float  gr[3][16][16];     //  3 KB  group gates (real)
    __shared__ float  gi[3][16][16];     //  3 KB  group gates (imag)
    __shared__ float2 Uq[N_QUBITS][2][2];
    __shared__ float  red[256];

    const int tid = threadIdx.x;
    const int b   = blockIdx.x;
    const int ln  = tid & 31;   // lane within wave32
    const int w   = tid >> 5;   // wave id (0..7)

    // ---- init |0...0> ----
    #pragma unroll
    for (int k = 0; k < 16; ++k) {
        sre[tid * 16 + k] = 0.f;
        sim_[tid * 16 + k] = 0.f;
    }
    if (tid == 0) sre[0] = 1.f;
    __syncthreads();

    const float xv = x[b];
    const float x1 = asinf(xv);
    const float x2 = acosf(xv * xv);

    for (int l = 0; l < N_LAYERS; ++l) {
        // ---- fused per-qubit 2x2 gate: RX(t2) RZ(t1) RX(t0) RZ(x2) RY(x1) ----
        if (tid < N_QUBITS) {
            const int q = tid;
            float c = cosf(0.5f * x1), s = sinf(0.5f * x1);
            C2x2 M;
            M.m00 = make_float2(c, 0.f);  M.m01 = make_float2(-s, 0.f);
            M.m10 = make_float2(s, 0.f);  M.m11 = make_float2(c, 0.f);
            lmul_rz(x2, M);
            const float t0 = params[q * 12 + l * 3 + 0];
            const float t1 = params[q * 12 + l * 3 + 1];
            const float t2 = params[q * 12 + l * 3 + 2];
            lmul_rx(t0, M); lmul_rz(t1, M); lmul_rx(t2, M);
            Uq[q][0][0] = M.m00; Uq[q][0][1] = M.m01;
            Uq[q][1][0] = M.m10; Uq[q][1][1] = M.m11;
        }
        __syncthreads();

        // ---- 16x16 Kronecker gates, one entry per thread per group ----
        {
            const int i = tid >> 4, m = tid & 15;
            #pragma unroll
            for (int g = 0; g < 3; ++g) {
                float2 p = Uq[4 * g + 0][(i >> 3) & 1][(m >> 3) & 1];
                p = cmul(p, Uq[4 * g + 1][(i >> 2) & 1][(m >> 2) & 1]);
                p = cmul(p, Uq[4 * g + 2][(i >> 1) & 1][(m >> 1) & 1]);
                p = cmul(p, Uq[4 * g + 3][i & 1][m & 1]);
                gr[g][i][m] = p.x;
                gi[g][i][m] = p.y;
            }
        }
        __syncthreads();

        // ---- apply the three 16x16 group gates via f32 WMMA ----
        for (int g = 0; g < 3; ++g) {
            #pragma unroll
            for (int tt = 0; tt < 2; ++tt) {
                const int t     = w + tt * 8;        // tile (16 state columns)
                const int col   = t * 16 + (ln & 15);
                const int Mrow  = ln & 15;
                const int khalf = (ln & 16) ? 2 : 0;
                v8f accr = {};
                v8f acci = {};
                #pragma unroll
                for (int kb = 0; kb < 16; kb += 4) {
                    const int K0 = kb + khalf, K1 = K0 + 1;
                    v2f ar, ai, nai, br, bi;
                    ar[0] = gr[g][Mrow][K0];  ar[1] = gr[g][Mrow][K1];
                    ai[0] = gi[g][Mrow][K0];  ai[1] = gi[g][Mrow][K1];
                    nai = -ai;
                    const int A0 = gaddr(g, K0, col);
                    const int A1 = gaddr(g, K1, col);
                    br[0] = sre[A0];  br[1] = sre[A1];
                    bi[0] = sim_[A0]; bi[1] = sim_[A1];
                    // D_r += Gr*Sr - Gi*Si ; D_i += Gr*Si + Gi*Sr
                    accr = __builtin_amdgcn_wmma_f32_16x16x4_f32(
                        false, ar, false, br, (short)0, accr, false, false);
                    accr = __builtin_amdgcn_wmma_f32_16x16x4_f32(
                        false, nai, false, bi, (short)0, accr, false, false);
                    acci = __builtin_amdgcn_wmma_f32_16x16x4_f32(
                        false, ar, false, bi, (short)0, acci, false, false);
                    acci = __builtin_amdgcn_wmma_f32_16x16x4_f32(
                        false, ai, false, br, (short)0, acci, false, false);
                }
                const int rbase = (ln & 16) ? 8 : 0;
                #pragma unroll
                for (int v = 0; v < 8; ++v) {
                    const int A = gaddr(g, rbase + v, col);
                    sre[A]  = accr[v];
                    sim_[A] = acci[v];
                }
            }
            __syncthreads();
        }

        // ---- full CNOT chain == inverse-Gray-code permutation ----
        float pr[16], pi[16];
        #pragma unroll
        for (int k = 0; k < 16; ++k) {
            const int j   = tid * 16 + k;
            const int src = j ^ (j >> 1);
            pr[k] = sre[src];
            pi[k] = sim_[src];
        }
        __syncthreads();
        #pragma unroll
        for (int k = 0; k < 16; ++k) {
            sre[tid * 16 + k]  = pr[k];
            sim_[tid * 16 + k] = pi[k];
        }
        __syncthreads();
    }

    // ---- <Z on qubit 0> = sum sign(MSB) * |amp|^2 ----
    float p = 0.f;
    #pragma unroll
    for (int k = 0; k < 16; ++k) {
        const int j = tid * 16 + k;
        p += sre[j] * sre[j] + sim_[j] * sim_[j];
    }
    if (tid >= 128) p = -p;   // MSB==1 half
    red[tid] = p;
    __syncthreads();
    for (int s2 = 128; s2 > 0; s2 >>= 1) {
        if (tid < s2) red[tid] += red[tid + s2];
        __syncthreads();
    }
    if (tid == 0) out[b] = red[0];
}

extern "C" void kernel_launch(void* const* d_in, const int* in_sizes, int n_in,
                              void* d_out, int out_size, void* d_ws, size_t ws_size,
                              hipStream_t stream) {
    const float* x      = (const float*)d_in[0];   // (256,)  float32
    const float* params = (const float*)d_in[1];   // (144,)  float32
    float*       out    = (float*)d_out;           // (256,)  float32
    (void)in_sizes; (void)n_in; (void)d_ws; (void)ws_size;
    const int nb = out_size > 0 ? out_size : BATCH;
    qsim_hea_kernel<<<dim3(nb), dim3(256), 0, stream>>>(x, params, out);
}